// HashEncoding_16716012716752
// MI455X (gfx1250) — compile-verified
//
#include <hip/hip_runtime.h>
#include <math.h>
#include <stdint.h>

// ---------------------------------------------------------------------------
// Instant-NGP hash-grid encoding for MI455X (gfx1250, wave32).
//
// Layout strategy:
//   thread tid  ->  level = tid & 15, point = tid >> 4
//   A wave32 handles 2 points x 16 levels; the float2 result lands at
//   out2[tid] (out2 = (float2*)out), i.e. a perfectly linear, fully
//   coalesced 256B store per wave.
//
// Cache strategy (the MI455X-specific part):
//   - 64 MB hash table fits in the 192 MB L2 -> gathers use default RT
//     policy so the table stays L2-resident.
//   - 128 MB output is write-once streaming -> non-temporal stores
//     (TH_NT) so the output stream does not evict the hash table.
// ---------------------------------------------------------------------------

#define NUM_LEVELS 16
#define LOG2_HASHMAP_SIZE 19
#define TABLE_MASK ((1u << LOG2_HASHMAP_SIZE) - 1u)

typedef __attribute__((ext_vector_type(2))) float v2f;

struct Scalings { float s[NUM_LEVELS]; };

__global__ __launch_bounds__(256) void hashenc_kernel(
    const float* __restrict__ x,      // (N, 3) float32
    const v2f*   __restrict__ table,  // (TABLE_SIZE*NUM_LEVELS) float2
    v2f*         __restrict__ out,    // (N, 16) float2  == (N, 32) float
    Scalings sc, int n_threads)
{
    int tid = blockIdx.x * blockDim.x + threadIdx.x;
    if (tid >= n_threads) return;

    uint32_t level = (uint32_t)tid & (NUM_LEVELS - 1u);
    uint32_t point = (uint32_t)tid >> 4;

    // 16 consecutive lanes read the same point -> single merged L0 request.
    float px = x[point * 3 + 0];
    float py = x[point * 3 + 1];
    float pz = x[point * 3 + 2];

    float s  = sc.s[level];
    float sx = px * s, sy = py * s, sz = pz * s;

    float fx = floorf(sx), fy = floorf(sy), fz = floorf(sz);
    float ox = sx - fx,    oy = sy - fy,    oz = sz - fz;

    uint32_t ifx = (uint32_t)fx;
    uint32_t ify = (uint32_t)fy;
    uint32_t ifz = (uint32_t)fz;
    uint32_t icx = (uint32_t)ceilf(sx);
    uint32_t icy = (uint32_t)ceilf(sy);
    uint32_t icz = (uint32_t)ceilf(sz);

    // h = (x*1) ^ (y*2654435761) ^ (z*805459861), masked, + level offset
    uint32_t base = level << LOG2_HASHMAP_SIZE;
    uint32_t hyf = ify * 2654435761u, hyc = icy * 2654435761u;
    uint32_t hzf = ifz * 805459861u,  hzc = icz * 805459861u;

    // Corner order matches CORNER_MASK rows in the reference:
    //   0:(1,1,1) 1:(1,0,1) 2:(0,0,1) 3:(0,1,1)
    //   4:(1,1,0) 5:(1,0,0) 6:(0,0,0) 7:(0,1,0)
    uint32_t i0 = ((icx ^ hyc ^ hzc) & TABLE_MASK) + base;
    uint32_t i1 = ((icx ^ hyf ^ hzc) & TABLE_MASK) + base;
    uint32_t i2 = ((ifx ^ hyf ^ hzc) & TABLE_MASK) + base;
    uint32_t i3 = ((ifx ^ hyc ^ hzc) & TABLE_MASK) + base;
    uint32_t i4 = ((icx ^ hyc ^ hzf) & TABLE_MASK) + base;
    uint32_t i5 = ((icx ^ hyf ^ hzf) & TABLE_MASK) + base;
    uint32_t i6 = ((ifx ^ hyf ^ hzf) & TABLE_MASK) + base;
    uint32_t i7 = ((ifx ^ hyc ^ hzf) & TABLE_MASK) + base;

    // 8 independent global_load_b64 gathers (L2-resident table, RT policy).
    v2f f0 = table[i0];
    v2f f1 = table[i1];
    v2f f2 = table[i2];
    v2f f3 = table[i3];
    v2f f4 = table[i4];
    v2f f5 = table[i5];
    v2f f6 = table[i6];
    v2f f7 = table[i7];

    float rx = 1.0f - ox, ry = 1.0f - oy, rz = 1.0f - oz;

    v2f f03 = f0 * ox + f3 * rx;
    v2f f12 = f1 * ox + f2 * rx;
    v2f f56 = f5 * ox + f6 * rx;
    v2f f47 = f4 * ox + f7 * rx;

    v2f f0312 = f03 * oy + f12 * ry;
    v2f f4756 = f47 * oy + f56 * ry;

    v2f enc = f0312 * oz + f4756 * rz;

    // Streaming output: non-temporal so 128 MB of writes don't evict the
    // 64 MB hash table from L2.  Linear in tid -> fully coalesced.
    __builtin_nontemporal_store(enc, &out[tid]);
}

extern "C" void kernel_launch(void* const* d_in, const int* in_sizes, int n_in,
                              void* d_out, int out_size, void* d_ws, size_t ws_size,
                              hipStream_t stream) {
    const float* x     = (const float*)d_in[0];
    const v2f*   table = (const v2f*)d_in[1];
    v2f*         out   = (v2f*)d_out;

    int n_points  = in_sizes[0] / 3;
    int n_threads = n_points * NUM_LEVELS;

    // Reproduce NumPy float64 computation of SCALINGS exactly:
    //   growth = exp((log(4096) - log(16)) / 15); s[i] = floor(16 * growth**i)
    Scalings sc;
    double growth = exp((log(4096.0) - log(16.0)) / 15.0);
    for (int i = 0; i < NUM_LEVELS; ++i) {
        sc.s[i] = (float)floor(16.0 * pow(growth, (double)i));
    }

    int block = 256;
    int grid  = (n_threads + block - 1) / block;
    hashenc_kernel<<<grid, block, 0, stream>>>(x, table, out, sc, n_threads);
}